// DeformableConv_17300128268592
// MI455X (gfx1250) — compile-verified
//
#include <hip/hip_runtime.h>

typedef float v2f __attribute__((ext_vector_type(2)));
typedef float v8f __attribute__((ext_vector_type(8)));

#define Nn   2
#define Hh   96
#define Ww   160
#define Cc   32
#define OHh  94
#define OWw  158
#define NPIX (Nn * OHh * OWw)   // 29704
#define OCH  36                 // offset channels
#define WRED_SZ 4608            // 16 f * 9 k9 * 16 c16 * 2 g
#define CB_OFF   WRED_SZ        // 16 floats
#define BFRAG_OFF 4624          // float2[72*3*32] = 6912 v2f = 13824 floats
#define BFRAG_N  6912           // number of float2 elements

// ---------------------------------------------------------------------------
// Prep:
//  W_red[f*288 + k9*32 + 2*c16 + g] = sum_{cl<16} ck[(k9*16 + 2j+g)*512 + f*32 + ch*16 + cl]
//  CB[f] = sum_{c<32} cb[f*32+c]
//  Bfrag[(s*3+t)*32 + lane] = { B[k][col], B[k+1][col] }  (fragment-order, zero-padded)
//     with k = s*4 + 2*(lane>>4), col = (lane&15) + 16*t, B[k][col] = okern[k*36+col]
// ---------------------------------------------------------------------------
__global__ __launch_bounds__(256) void prep_kernel(const float* __restrict__ ck,
                                                   const float* __restrict__ cb,
                                                   const float* __restrict__ okern,
                                                   float* __restrict__ ws) {
    const int tid = blockIdx.x * blockDim.x + threadIdx.x;
    const int nthreads = gridDim.x * blockDim.x;

    for (int idx = tid; idx < WRED_SZ; idx += nthreads) {
        int f   = idx / 288;
        int rem = idx % 288;
        int k9  = rem / 32;
        int r2  = rem % 32;
        int c16 = r2 >> 1;
        int g   = r2 & 1;
        int ch  = c16 >> 3;
        int j   = c16 & 7;
        int i   = 2 * j + g;
        const float* src = ck + (k9 * 16 + i) * 512 + f * 32 + ch * 16;
        float s = 0.f;
        #pragma unroll
        for (int cl = 0; cl < 16; ++cl) s += src[cl];
        ws[idx] = s;
    }
    if (tid < 16) {
        float s = 0.f;
        #pragma unroll
        for (int c = 0; c < 32; ++c) s += cb[tid * 32 + c];
        ws[CB_OFF + tid] = s;
    }
    for (int idx = tid; idx < BFRAG_N; idx += nthreads) {
        int s    = idx / 96;
        int r    = idx % 96;
        int t    = r / 32;
        int lane = r % 32;
        int kl   = lane >> 4;
        int m    = lane & 15;
        int k    = s * 4 + 2 * kl;
        int col  = m + 16 * t;
        float b0 = (col < OCH) ? okern[k * OCH + col] : 0.f;
        float b1 = (col < OCH) ? okern[(k + 1) * OCH + col] : 0.f;
        ws[BFRAG_OFF + 2 * idx]     = b0;
        ws[BFRAG_OFF + 2 * idx + 1] = b1;
    }
}

// ---------------------------------------------------------------------------
// Fused: offsets conv (WMMA f32 16x16x4) -> bilinear sample -> reduced conv
// One wave (32 threads) per 16-pixel tile.
// ---------------------------------------------------------------------------
__global__ __launch_bounds__(32) void deform_kernel(const float* __restrict__ vol,
                                                    const float* __restrict__ obias,
                                                    const float* __restrict__ ws,
                                                    float* __restrict__ out) {
    __shared__ float ldsW[WRED_SZ];
    __shared__ float ldsCB[16];
    __shared__ float ldsOff[16 * 40];   // [pixel][offset channel], padded

    const int lane = threadIdx.x;       // 0..31
    const int m    = lane & 15;
    const int kl   = lane >> 4;

    // Stage reduced weights into LDS (b128 loads/stores).
    {
        const float4* src = (const float4*)ws;
        float4*       dst = (float4*)ldsW;
        #pragma unroll
        for (int idx = lane; idx < WRED_SZ / 4; idx += 32) dst[idx] = src[idx];
        if (lane < 16) ldsCB[lane] = ws[CB_OFF + lane];
    }

    // ---------------- Phase 1: offsets conv via V_WMMA_F32_16X16X4_F32 ------
    // M = 16 pixels, N = 36 channels (3 N-tiles padded to 48), K = 288.
    const int P0 = blockIdx.x * 16;
    int Pm = P0 + m; if (Pm >= NPIX) Pm = NPIX - 1;     // clamp ragged tile
    const int na = Pm / (OHh * OWw);
    const int ra = Pm % (OHh * OWw);
    const int ya = ra / OWw;
    const int xa = ra % OWw;

    // Lane's A elements for step s: A[m][s*4 + 2*kl + r], r = 0,1
    const float* pbase = vol + na * (Hh * Ww * Cc) + (ya * Ww + xa) * Cc + 2 * kl;
    const v2f*   bfl   = ((const v2f*)(ws + BFRAG_OFF)) + lane;

    v8f acc0 = {}; v8f acc1 = {}; v8f acc2 = {};

    #pragma unroll
    for (int kh = 0; kh < 3; ++kh) {
        #pragma unroll
        for (int kw = 0; kw < 3; ++kw) {
            const float* ap = pbase + (kh * Ww + kw) * Cc;
            #pragma unroll
            for (int t8 = 0; t8 < 8; ++t8) {
                const int s = (kh * 3 + kw) * 8 + t8;
                v2f a  = *(const v2f*)(ap + t8 * 4);
                v2f b0 = bfl[(s * 3 + 0) * 32];
                v2f b1 = bfl[(s * 3 + 1) * 32];
                v2f b2 = bfl[(s * 3 + 2) * 32];
                acc0 = __builtin_amdgcn_wmma_f32_16x16x4_f32(false, a, false, b0, (short)0, acc0, false, false);
                acc1 = __builtin_amdgcn_wmma_f32_16x16x4_f32(false, a, false, b1, (short)0, acc1, false, false);
                acc2 = __builtin_amdgcn_wmma_f32_16x16x4_f32(false, a, false, b2, (short)0, acc2, false, false);
            }
        }
    }

    // D layout: lane col o = (lane&15)+16t, rows (pixels) r + 8*(lane>>4)
    {
        const int o0 = m, o1 = m + 16, o2 = m + 32;
        const float bo0 = obias[o0];
        const float bo1 = obias[o1];
        const float bo2 = (o2 < OCH) ? obias[o2] : 0.f;
        #pragma unroll
        for (int r = 0; r < 8; ++r) {
            const int pix = r + 8 * kl;
            ldsOff[pix * 40 + o0] = acc0[r] + bo0;
            ldsOff[pix * 40 + o1] = acc1[r] + bo1;
            if (o2 < OCH) ldsOff[pix * 40 + o2] = acc2[r] + bo2;
        }
    }
    __syncthreads();

    // ---------------- Phase 2: bilinear sample + reduced final conv ---------
    // lane = (pixel p, half h): half h owns source channels [16h,16h+16)
    // and output filters [8h, 8h+8) (band structure of the grouped conv).
    const int p = m;
    const int h = kl;
    const int P = P0 + p;
    if (P < NPIX) {
        const int n  = P / (OHh * OWw);
        const int rr = P % (OHh * OWw);
        const int y  = rr / OWw;
        const int x  = rr % OWw;

        float acc[8];
        #pragma unroll
        for (int ff = 0; ff < 8; ++ff) acc[ff] = ldsCB[h * 8 + ff];

        const float* voln = vol + n * (Hh * Ww * Cc);
        const float  yc   = (float)(y + 1);          // absolute row (pad_y = 1)

        for (int k9 = 0; k9 < 9; ++k9) {
            const int fh = k9 / 3, fw = k9 % 3;
            #pragma unroll
            for (int g = 0; g < 2; ++g) {
                const float dy = ldsOff[p * 40 + k9 * 4 + g];
                const float dx = ldsOff[p * 40 + k9 * 4 + 2 + g];
                const float rx = (float)(fh - 1) + dy;   // kernel_vs + dy
                const float ry = (float)(fw - 1) + dx;   // kernel_us + dx
                int x0 = (int)rx, y0 = (int)ry;          // trunc toward zero
                int x1 = x0 + 1,  y1 = y0 + 1;
                y0 = y0 < 0 ? 0 : (y0 > Hh - 1 ? Hh - 1 : y0);
                y1 = y1 < 0 ? 0 : (y1 > Hh - 1 ? Hh - 1 : y1);
                x0 = x0 < 0 ? 0 : (x0 > Ww - 1 ? Ww - 1 : x0);
                x1 = x1 < 0 ? 0 : (x1 > Ww - 1 ? Ww - 1 : x1);
                const float w0 = ((float)y1 - ry) * ((float)x1 - rx);
                const float w1 = ((float)y1 - yc) * (rx - (float)x0);
                const float w2 = (ry - (float)y0) * ((float)x1 - rx);
                const float w3 = (ry - (float)y0) * (rx - (float)x0);
                const float4* q0 = (const float4*)(voln + (y0 * Ww + x0) * Cc + h * 16);
                const float4* q1 = (const float4*)(voln + (y0 * Ww + x1) * Cc + h * 16);
                const float4* q2 = (const float4*)(voln + (y1 * Ww + x0) * Cc + h * 16);
                const float4* q3 = (const float4*)(voln + (y1 * Ww + x1) * Cc + h * 16);
                const float* wrow = &ldsW[(h * 8) * 288 + k9 * 32 + g];
                #pragma unroll
                for (int q = 0; q < 4; ++q) {
                    const float4 a0 = q0[q], a1 = q1[q], a2 = q2[q], a3 = q3[q];
                    const float vv[4] = {
                        a0.x * w0 + a1.x * w1 + a2.x * w2 + a3.x * w3,
                        a0.y * w0 + a1.y * w1 + a2.y * w2 + a3.y * w3,
                        a0.z * w0 + a1.z * w1 + a2.z * w2 + a3.z * w3,
                        a0.w * w0 + a1.w * w1 + a2.w * w2 + a3.w * w3 };
                    #pragma unroll
                    for (int e = 0; e < 4; ++e) {
                        const int c16 = q * 4 + e;
                        #pragma unroll
                        for (int ff = 0; ff < 8; ++ff)
                            acc[ff] = fmaf(vv[e], wrow[ff * 288 + 2 * c16], acc[ff]);
                    }
                }
            }
        }
        float* op = out + P * 16 + h * 8;
        #pragma unroll
        for (int ff = 0; ff < 8; ++ff) op[ff] = acc[ff];
    }
}

// ---------------------------------------------------------------------------
extern "C" void kernel_launch(void* const* d_in, const int* in_sizes, int n_in,
                              void* d_out, int out_size, void* d_ws, size_t ws_size,
                              hipStream_t stream) {
    const float* volume        = (const float*)d_in[0];
    const float* offset_kernel = (const float*)d_in[1];
    const float* offset_bias   = (const float*)d_in[2];
    const float* conv_kernel   = (const float*)d_in[3];
    const float* conv_bias     = (const float*)d_in[4];
    float*       out           = (float*)d_out;
    float*       ws            = (float*)d_ws;   // 4624 + 13824 floats (~74 KB)

    prep_kernel<<<28, 256, 0, stream>>>(conv_kernel, conv_bias, offset_kernel, ws);

    const int ntiles = (NPIX + 15) / 16;         // 1857
    deform_kernel<<<ntiles, 32, 0, stream>>>(volume, offset_bias, ws, out);
}